// SharedDecHead_53850299957575
// MI455X (gfx1250) — compile-verified
//
#include <hip/hip_runtime.h>
#include <math.h>

#define BB 4
#define NQ 1000
#define CH 256
#define HH 8
#define DHD 32
#define LL 4
#define PP 4
#define KN 10
#define FFD 1024
#define LIN_T 19947
#define NCLS 2

typedef __attribute__((ext_vector_type(2))) float v2f;
typedef __attribute__((ext_vector_type(8))) float v8f;

// ---------------------------------------------------------------------------
// fp32 GEMM: Y[M x N] = act(X[M x K] @ W[K x N] + bias) via V_WMMA_F32_16X16X4.
// Block = 128 threads (4 waves). Block tile: 16 (M) x 64 (N); wave w owns the
// 16x16 subtile at column w*16. Per KC=64 chunk:
//   - X chunk (16x64) staged in LDS via coalesced b128 loads (stride 68).
//   - W chunk (64x64) staged in LDS TRANSPOSED (Ws[n][k], stride 68) via
//     coalesced b128 global loads + scatter stores. This makes B-fragment
//     reads identical in form to A-fragment reads: contiguous 8B pairs at
//     immediate offsets -> ds_load_2addr_b64, no per-step address math.
//   - Inner loop: 16x { 2 ds loads + v_wmma_f32_16x16x4_f32 }, fully unrolled.
// Fragment-read bank mapping (stride 68): bank = (4*lane + k) mod 64 ->
// conflict-free for both half-groups.
// M edge handled by clamping load rows (garbage feeds only unstored rows).
// A frag (16x4): lanes 0-15 hold K={0,1}, lanes 16-31 hold K={2,3}.
// C/D: VGPR r -> rows r (lanes 0-15) / r+8 (lanes 16-31).
// ---------------------------------------------------------------------------
#define KC 64
#define XS_LD 68
#define WS_LD 68

__global__ __launch_bounds__(128) void gemm16_wmma(
    const float* __restrict__ X, const float* __restrict__ W,
    const float* __restrict__ bias, float* __restrict__ Y,
    int M, int K, int N, int relu)
{
  __shared__ float Xs[16 * XS_LD];
  __shared__ float Ws[64 * WS_LD];
  const int t    = threadIdx.x;
  const int wave = t >> 5;
  const int lane = t & 31;
  const int half = lane >> 4;
  const int lidx = lane & 15;
  const int colBase = blockIdx.x * 64;
  const int col  = colBase + wave * 16 + lidx;
  const int mbase = blockIdx.y * 16;
  // cooperative staging coords: 16-thread groups load float4 rows
  const int srow = t >> 4;          // 0..7
  const int sq   = (t & 15) * 4;    // 0,4,...,60
  // per-lane fragment base pointers (immediate offsets in the hot loop)
  const float* ApL = Xs + lidx * XS_LD + 2 * half;
  const float* BpL = Ws + (wave * 16 + lidx) * WS_LD + 2 * half;

  v8f acc = {};
  for (int k0 = 0; k0 < K; k0 += KC) {
    __syncthreads();
    // ---- stage X (16 rows x 64 k) ----
#pragma unroll
    for (int pp = 0; pp < 2; ++pp) {
      const int r  = srow + pp * 8;
      const int rc = min(mbase + r, M - 1);        // branch-free edge handling
      *(float4*)(Xs + r * XS_LD + sq) =
          *(const float4*)(X + (size_t)rc * K + k0 + sq);
    }
    // ---- stage W transposed (64 k x 64 n -> Ws[n][k]) ----
#pragma unroll
    for (int pp = 0; pp < 8; ++pp) {
      const int kr = srow + pp * 8;                // 0..63
      const float4 wv =
          *(const float4*)(W + (size_t)(k0 + kr) * N + colBase + sq);
      Ws[(sq + 0) * WS_LD + kr] = wv.x;
      Ws[(sq + 1) * WS_LD + kr] = wv.y;
      Ws[(sq + 2) * WS_LD + kr] = wv.z;
      Ws[(sq + 3) * WS_LD + kr] = wv.w;
    }
    __syncthreads();
    // ---- 16 WMMAs, pure LDS-fed ----
#pragma unroll
    for (int kk = 0; kk < KC; kk += 4) {
      v2f a, b;
      a.x = ApL[kk];
      a.y = ApL[kk + 1];
      b.x = BpL[kk];
      b.y = BpL[kk + 1];
      acc = __builtin_amdgcn_wmma_f32_16x16x4_f32(false, a, false, b,
                                                  (short)0, acc, false, false);
    }
  }
  const float bval = bias ? bias[col] : 0.0f;
#pragma unroll
  for (int r = 0; r < 8; ++r) {
    const int m = mbase + r + 8 * half;
    if (m < M) {
      float y = acc[r] + bval;
      if (relu) y = fmaxf(y, 0.0f);
      Y[(size_t)m * N + col] = y;
    }
  }
}

// ---------------------------------------------------------------------------
// maskf[b,n] = (float)cur_mask[b,n]
// ---------------------------------------------------------------------------
__global__ void make_maskf(const int* __restrict__ cm, float* __restrict__ mf) {
  const int i = blockIdx.x * blockDim.x + threadIdx.x;
  if (i < BB * NQ) mf[i] = (float)cm[i];
}

// Y = (A (+ Bp)) (* mask-broadcast-over-C)
__global__ void add_mask_ew(const float* __restrict__ A, const float* __restrict__ Bp,
                            const float* __restrict__ mf, float* __restrict__ Y,
                            int total) {
  const int i = blockIdx.x * blockDim.x + threadIdx.x;
  if (i >= total) return;
  float v = A[i];
  if (Bp) v += Bp[i];
  if (mf) v *= mf[i / CH];
  Y[i] = v;
}

// ---------------------------------------------------------------------------
// Top-K=10 nearest alive boxes per query (stable ties: lower index first,
// matching jax.lax.top_k on -d2). One thread per (b,n).
// ---------------------------------------------------------------------------
__global__ void topk_kernel(const float* __restrict__ pb,
                            const float* __restrict__ mf,
                            int* __restrict__ nb) {
  const int row = blockIdx.x * blockDim.x + threadIdx.x;
  if (row >= BB * NQ) return;
  const int b = row / NQ;
  const float cx = pb[row * 4 + 0], cy = pb[row * 4 + 1];
  float bd[KN]; int bi[KN];
#pragma unroll
  for (int k = 0; k < KN; ++k) { bd[k] = 3.0e30f; bi[k] = 0; }
  for (int j = 0; j < NQ; ++j) {
    const float dx = pb[(b * NQ + j) * 4 + 0] - cx;
    const float dy = pb[(b * NQ + j) * 4 + 1] - cy;
    float d = dx * dx + dy * dy;
    if (mf[b * NQ + j] < 0.5f) d += 1.0e9f;
    if (d < bd[KN - 1]) {
      int i = KN - 1;
      while (i > 0 && d < bd[i - 1]) { bd[i] = bd[i - 1]; bi[i] = bi[i - 1]; --i; }
      bd[i] = d; bi[i] = j;
    }
  }
#pragma unroll
  for (int k = 0; k < KN; ++k) nb[row * KN + k] = bi[k];
}

// ---------------------------------------------------------------------------
// rel[b,n,c] = max_k relu( relu(geo_k @ Wg)[c] + fW[b, nb_k, c] )
// ---------------------------------------------------------------------------
__global__ __launch_bounds__(CH) void rel_max_kernel(
    const float* __restrict__ pb, const int* __restrict__ nb,
    const float* __restrict__ fW, const float* __restrict__ Wg,
    float* __restrict__ rel) {
  const int row = blockIdx.x;
  const int c = threadIdx.x;
  const int b = row / NQ;
  const float cx = pb[row * 4 + 0], cy = pb[row * 4 + 1];
  const float wi = fmaxf(pb[row * 4 + 2], 1e-3f);
  const float hi = fmaxf(pb[row * 4 + 3], 1e-3f);
  const float w0 = Wg[0 * CH + c], w1 = Wg[1 * CH + c];
  const float w2 = Wg[2 * CH + c], w3 = Wg[3 * CH + c];
  float best = -3.0e38f;
#pragma unroll
  for (int k = 0; k < KN; ++k) {
    const int j = nb[row * KN + k];
    const float* q = pb + (size_t)(b * NQ + j) * 4;
    const float dx = (q[0] - cx) / wi;
    const float dy = (q[1] - cy) / hi;
    const float dw = logf(fmaxf(q[2], 1e-3f) / wi);
    const float dh = logf(fmaxf(q[3], 1e-3f) / hi);
    float g = dx * w0 + dy * w1 + dw * w2 + dh * w3;
    g = fmaxf(g, 0.0f);
    const float v = fmaxf(g + fW[(size_t)(b * NQ + j) * CH + c], 0.0f);
    best = fmaxf(best, v);
  }
  rel[(size_t)row * CH + c] = best;
}

// ---------------------------------------------------------------------------
// Y = LayerNorm( (A (+ Bp)) (* mask), gamma, beta )  — two-pass mean/var
// ---------------------------------------------------------------------------
__global__ __launch_bounds__(CH) void ln_res_kernel(
    const float* __restrict__ A, const float* __restrict__ Bp,
    const float* __restrict__ mf, const float* __restrict__ g,
    const float* __restrict__ be, float* __restrict__ Y) {
  __shared__ float sm[CH];
  const int row = blockIdx.x;
  const int c = threadIdx.x;
  float v = A[(size_t)row * CH + c];
  if (Bp) v += Bp[(size_t)row * CH + c];
  if (mf) v *= mf[row];
  sm[c] = v; __syncthreads();
  for (int st = CH / 2; st > 0; st >>= 1) {
    if (c < st) sm[c] += sm[c + st];
    __syncthreads();
  }
  const float mean = sm[0] * (1.0f / CH);
  __syncthreads();
  const float dv = v - mean;
  sm[c] = dv * dv; __syncthreads();
  for (int st = CH / 2; st > 0; st >>= 1) {
    if (c < st) sm[c] += sm[c + st];
    __syncthreads();
  }
  const float var = sm[0] * (1.0f / CH);
  Y[(size_t)row * CH + c] = dv * rsqrtf(var + 1e-5f) * g[c] + be[c];
}

// ---------------------------------------------------------------------------
// K-neighbor multi-head attention. Block = (32,8): wave per head, lane per dim.
// ---------------------------------------------------------------------------
__global__ __launch_bounds__(256) void self_attn_kernel(
    const float* __restrict__ qh, const float* __restrict__ kh,
    const float* __restrict__ vh, const int* __restrict__ nb,
    const float* __restrict__ mf, float* __restrict__ outp) {
  const int row = blockIdx.x;
  const int h = threadIdx.y, d = threadIdx.x;
  const int b = row / NQ;
  const float qv = qh[(size_t)row * CH + h * DHD + d];
  int idx[KN]; float sc[KN];
#pragma unroll
  for (int k = 0; k < KN; ++k) {
    const int j = nb[row * KN + k];
    const int r2 = b * NQ + j;
    idx[k] = r2;
    float p = qv * kh[(size_t)r2 * CH + h * DHD + d];
#pragma unroll
    for (int m = 16; m >= 1; m >>= 1) p += __shfl_xor(p, m, 32);
    p *= 0.17677669529663689f;  // 1/sqrt(32)
    sc[k] = (mf[r2] > 0.5f) ? p : -1.0e9f;
  }
  float mx = sc[0];
#pragma unroll
  for (int k = 1; k < KN; ++k) mx = fmaxf(mx, sc[k]);
  float w[KN]; float ssum = 0.0f;
#pragma unroll
  for (int k = 0; k < KN; ++k) { w[k] = expf(sc[k] - mx); ssum += w[k]; }
  const float inv = 1.0f / ssum;
  float o = 0.0f;
#pragma unroll
  for (int k = 0; k < KN; ++k)
    o += w[k] * inv * vh[(size_t)idx[k] * CH + h * DHD + d];
  outp[(size_t)row * CH + h * DHD + d] = o;
}

// softmax over 16 sampling weights per (b,n,h), in place
__global__ void aw_softmax_kernel(float* __restrict__ aw) {
  const int i = blockIdx.x * blockDim.x + threadIdx.x;
  if (i >= BB * NQ * HH) return;
  float* p = aw + (size_t)i * 16;
  float mx = p[0];
#pragma unroll
  for (int j = 1; j < 16; ++j) mx = fmaxf(mx, p[j]);
  float e[16]; float s = 0.0f;
#pragma unroll
  for (int j = 0; j < 16; ++j) { e[j] = expf(p[j] - mx); s += e[j]; }
  const float inv = 1.0f / s;
#pragma unroll
  for (int j = 0; j < 16; ++j) p[j] = e[j] * inv;
}

__device__ __forceinline__ float samp1(const float* __restrict__ vb,
                                       int yi, int xi, int Hl, int Wl) {
  const bool inb = (yi >= 0) && (yi < Hl) && (xi >= 0) && (xi < Wl);
  const int yc = min(max(yi, 0), Hl - 1);
  const int xc = min(max(xi, 0), Wl - 1);
  const float v = vb[(size_t)(yc * Wl + xc) * CH];
  return inb ? v : 0.0f;
}

// ---------------------------------------------------------------------------
// Deformable bilinear sampling over 4 levels. Block=(32,8): head/dim layout.
// ---------------------------------------------------------------------------
__global__ __launch_bounds__(256) void deform_kernel(
    const float* __restrict__ val, const float* __restrict__ off,
    const float* __restrict__ aw, const float* __restrict__ refp,
    const float* __restrict__ mf, float* __restrict__ samp) {
  const int row = blockIdx.x;
  const int h = threadIdx.y, d = threadIdx.x;
  const int b = row / NQ;
  const float m = mf[row];
  const float rx = refp[row * 4 + 0] * m, ry = refp[row * 4 + 1] * m;
  const float rw = refp[row * 4 + 2] * m, rh = refp[row * 4 + 3] * m;
  const int Hls[LL] = {100, 50, 25, 13};
  const int Wls[LL] = {150, 75, 38, 19};
  const int sts[LL] = {0, 15000, 18750, 19700};
  float acc = 0.0f;
#pragma unroll
  for (int l = 0; l < LL; ++l) {
    const int Hl = Hls[l], Wl = Wls[l];
    const float* vb = val + (size_t)(b * LIN_T + sts[l]) * CH + h * DHD + d;
#pragma unroll
    for (int p = 0; p < PP; ++p) {
      const int oi = row * CH + ((h * LL + l) * PP + p) * 2;
      const float lx = rx + off[oi + 0] * (0.5f / PP) * rw;
      const float ly = ry + off[oi + 1] * (0.5f / PP) * rh;
      const float x = lx * Wl - 0.5f, y = ly * Hl - 0.5f;
      const float fx = floorf(x), fy = floorf(y);
      const int x0 = (int)fx, y0 = (int)fy;
      const float wx = x - fx, wy = y - fy;
      const float a = aw[(size_t)(row * HH + h) * 16 + l * PP + p];
      float s = samp1(vb, y0,     x0,     Hl, Wl) * (1.0f - wx) * (1.0f - wy)
              + samp1(vb, y0,     x0 + 1, Hl, Wl) * wx * (1.0f - wy)
              + samp1(vb, y0 + 1, x0,     Hl, Wl) * (1.0f - wx) * wy
              + samp1(vb, y0 + 1, x0 + 1, Hl, Wl) * wx * wy;
      acc += a * s;
    }
  }
  samp[(size_t)row * CH + h * DHD + d] = acc;
}

// ---------------------------------------------------------------------------
// Final heads: logits, boxes, masks. One thread per (b,n).
// ---------------------------------------------------------------------------
__device__ __forceinline__ float sigm(float x) { return 1.0f / (1.0f + expf(-x)); }

__global__ void final_head_kernel(
    const float* __restrict__ t4, const float* __restrict__ bh2,
    const float* __restrict__ Wc, const float* __restrict__ bc,
    const float* __restrict__ Wb3, const float* __restrict__ bb3,
    const float* __restrict__ refp, const float* __restrict__ prev_logits,
    const int* __restrict__ cur_mask, const int* __restrict__ gather_mask,
    const float* __restrict__ mf,
    float* __restrict__ out_logits, float* __restrict__ out_boxes,
    float* __restrict__ out_nmask, float* __restrict__ out_gmask) {
  const int row = blockIdx.x * blockDim.x + threadIdx.x;
  if (row >= BB * NQ) return;
  float lg[NCLS];
#pragma unroll
  for (int j = 0; j < NCLS; ++j) lg[j] = bc[j];
  float y4[4];
#pragma unroll
  for (int j = 0; j < 4; ++j) y4[j] = bb3[j];
  for (int c = 0; c < CH; ++c) {
    const float tv = t4[(size_t)row * CH + c];
    const float bv = bh2[(size_t)row * CH + c];
#pragma unroll
    for (int j = 0; j < NCLS; ++j) lg[j] += tv * Wc[c * NCLS + j];
#pragma unroll
    for (int j = 0; j < 4; ++j) y4[j] += bv * Wb3[c * 4 + j];
  }
#pragma unroll
  for (int j = 0; j < 4; ++j) {
    float r = refp[row * 4 + j];
    r = fminf(fmaxf(r, 0.0f), 1.0f);
    const float is = logf(fmaxf(r, 1e-5f) / fmaxf(1.0f - r, 1e-5f));
    out_boxes[row * 4 + j] = sigm(y4[j] + is);
  }
  float maxs = sigm(lg[0]);
#pragma unroll
  for (int j = 1; j < NCLS; ++j) maxs = fmaxf(maxs, sigm(lg[j]));
  const float alive = (maxs >= 0.5f) ? 1.0f : 0.0f;
  out_nmask[row] = (mf[row] - alive) * ((maxs >= 0.0f) ? 1.0f : 0.0f);
  out_gmask[row] = (float)gather_mask[row] + alive;
#pragma unroll
  for (int j = 0; j < NCLS; ++j)
    out_logits[row * NCLS + j] =
        (cur_mask[row] > 0) ? lg[j] : prev_logits[row * NCLS + j];
}

// ---------------------------------------------------------------------------
extern "C" void kernel_launch(void* const* d_in, const int* in_sizes, int n_in,
                              void* d_out, int out_size, void* d_ws, size_t ws_size,
                              hipStream_t stream) {
  (void)in_sizes; (void)n_in; (void)out_size; (void)ws_size;
  const float* tgt   = (const float*)d_in[0];
  const float* qpos  = (const float*)d_in[1];
  const float* refp  = (const float*)d_in[2];
  const float* src   = (const float*)d_in[3];
  const float* pb    = (const float*)d_in[4];
  const float* prevl = (const float*)d_in[5];
  const int*   cmI   = (const int*)d_in[6];
  const int*   gmI   = (const int*)d_in[7];
  const float* Wg    = (const float*)d_in[10];
  const float* Wf_rel= (const float*)d_in[11];
  const float* Wo_rel= (const float*)d_in[12];
  const float* g_rel = (const float*)d_in[13];
  const float* b_rel = (const float*)d_in[14];
  const float* Wq    = (const float*)d_in[15];
  const float* Wk    = (const float*)d_in[16];
  const float* Wv    = (const float*)d_in[17];
  const float* Wo    = (const float*)d_in[18];
  const float* W_off = (const float*)d_in[19];
  const float* b_off = (const float*)d_in[20];
  const float* W_attw= (const float*)d_in[21];
  const float* b_attw= (const float*)d_in[22];
  const float* W_val = (const float*)d_in[23];
  const float* W_outc= (const float*)d_in[24];
  const float* g1    = (const float*)d_in[25];
  const float* bn1   = (const float*)d_in[26];
  const float* g2    = (const float*)d_in[27];
  const float* bn2   = (const float*)d_in[28];
  const float* W1    = (const float*)d_in[29];
  const float* b1    = (const float*)d_in[30];
  const float* W2    = (const float*)d_in[31];
  const float* b2    = (const float*)d_in[32];
  const float* g3    = (const float*)d_in[33];
  const float* bn3   = (const float*)d_in[34];
  const float* Wc    = (const float*)d_in[35];
  const float* bc    = (const float*)d_in[36];
  const float* Wb1   = (const float*)d_in[37];
  const float* bb1   = (const float*)d_in[38];
  const float* Wb2   = (const float*)d_in[39];
  const float* bb2   = (const float*)d_in[40];
  const float* Wb3   = (const float*)d_in[41];
  const float* bb3   = (const float*)d_in[42];

  const size_t BN = (size_t)BB * NQ;
  const size_t BNC = BN * CH;
  float* ws = (float*)d_ws;
  size_t o = 0;
  int* nb       = (int*)(ws + o); o += BN * KN;
  float* maskf  = ws + o; o += BN;
  float* f      = ws + o; o += BNC;
  float* fW     = ws + o; o += BNC;
  float* rel    = ws + o; o += BNC;
  float* relW   = ws + o; o += BNC;
  float* t1     = ws + o; o += BNC;
  float* qk     = ws + o; o += BNC;
  float* qh     = ws + o; o += BNC;
  float* kh     = ws + o; o += BNC;
  float* vh     = ws + o; o += BNC;
  float* attno  = ws + o; o += BNC;
  float* lo     = ws + o; o += BNC;
  float* t2     = ws + o; o += BNC;
  float* q      = ws + o; o += BNC;
  float* off    = ws + o; o += BNC;
  float* aw     = ws + o; o += BN * HH * 16;
  float* sampo  = ws + o; o += BNC;
  float* ca     = ws + o; o += BNC;
  float* t3     = ws + o; o += BNC;
  float* h1     = ws + o; o += BN * FFD;
  float* ffo    = ws + o; o += BNC;
  float* bh1    = ws + o; o += BNC;
  float* bh2    = ws + o; o += BNC;
  float* val    = ws + o; o += (size_t)BB * LIN_T * CH;

  float* out_t      = (float*)d_out;
  float* out_logits = out_t + BNC;
  float* out_boxes  = out_logits + BN * NCLS;
  float* out_nmask  = out_boxes + BN * 4;
  float* out_gmask  = out_nmask + BN;

  auto gemm = [&](const float* X, const float* W, const float* bias, float* Y,
                  int M, int K, int Nn, int relu) {
    dim3 grid(Nn / 64, (M + 15) / 16);
    gemm16_wmma<<<grid, dim3(128), 0, stream>>>(X, W, bias, Y, M, K, Nn, relu);
  };
  auto ln = [&](const float* A, const float* Bp, const float* mf_,
                const float* g, const float* be, float* Y) {
    ln_res_kernel<<<(int)BN, CH, 0, stream>>>(A, Bp, mf_, g, be, Y);
  };
  const int EW = (int)((BNC + 255) / 256);

  // ---- masks & masked features -------------------------------------------
  make_maskf<<<(int)((BN + 255) / 256), 256, 0, stream>>>(cmI, maskf);
  add_mask_ew<<<EW, 256, 0, stream>>>(tgt, nullptr, maskf, f, (int)BNC);

  // ---- relation module ----------------------------------------------------
  topk_kernel<<<(int)((BN + 255) / 256), 256, 0, stream>>>(pb, maskf, nb);
  gemm(f, Wf_rel, nullptr, fW, (int)BN, CH, CH, 0);   // gather commutes: f@Wf then gather
  rel_max_kernel<<<(int)BN, CH, 0, stream>>>(pb, nb, fW, Wg, rel);
  gemm(rel, Wo_rel, nullptr, relW, (int)BN, CH, CH, 0);
  ln(f, relW, nullptr, g_rel, b_rel, t1);

  // ---- K-neighbor self-attention -----------------------------------------
  add_mask_ew<<<EW, 256, 0, stream>>>(t1, qpos, nullptr, qk, (int)BNC);
  gemm(qk, Wq, nullptr, qh, (int)BN, CH, CH, 0);
  gemm(qk, Wk, nullptr, kh, (int)BN, CH, CH, 0);
  gemm(t1, Wv, nullptr, vh, (int)BN, CH, CH, 0);
  self_attn_kernel<<<(int)BN, dim3(32, 8), 0, stream>>>(qh, kh, vh, nb, maskf, attno);
  gemm(attno, Wo, nullptr, lo, (int)BN, CH, CH, 0);
  ln(t1, lo, maskf, g2, bn2, t2);

  // ---- deformable cross-attention ----------------------------------------
  add_mask_ew<<<EW, 256, 0, stream>>>(t2, qpos, maskf, q, (int)BNC);
  gemm(q, W_off, b_off, off, (int)BN, CH, HH * LL * PP * 2, 0);
  gemm(q, W_attw, b_attw, aw, (int)BN, CH, HH * LL * PP, 0);
  aw_softmax_kernel<<<(int)((BN * HH + 255) / 256), 256, 0, stream>>>(aw);
  gemm(src, W_val, nullptr, val, BB * LIN_T, CH, CH, 0);
  deform_kernel<<<(int)BN, dim3(32, 8), 0, stream>>>(val, off, aw, refp, maskf, sampo);
  gemm(sampo, W_outc, nullptr, ca, (int)BN, CH, CH, 0);
  ln(t2, ca, maskf, g1, bn1, t3);

  // ---- FFN ---------------------------------------------------------------
  gemm(t3, W1, b1, h1, (int)BN, CH, FFD, 1);
  gemm(h1, W2, b2, ffo, (int)BN, FFD, CH, 0);
  ln(t3, ffo, nullptr, g3, bn3, out_t);            // t4 -> first output

  // ---- heads -------------------------------------------------------------
  gemm(out_t, Wb1, bb1, bh1, (int)BN, CH, CH, 1);
  gemm(bh1, Wb2, bb2, bh2, (int)BN, CH, CH, 1);
  final_head_kernel<<<(int)((BN + 255) / 256), 256, 0, stream>>>(
      out_t, bh2, Wc, bc, Wb3, bb3, refp, prevl, cmI, gmI, maskf,
      out_logits, out_boxes, out_nmask, out_gmask);
}